// CRF_85169201480103
// MI455X (gfx1250) — compile-verified
//
#include <hip/hip_runtime.h>
#include <hip/hip_bf16.h>

// CRF NLL: B=2048, S=4096, T=3. Memory-bound: 168 MB @ 23.3 TB/s ~= 7.2 us floor.
// K1: per-(batch,chunk) thread composes a 3x3 transfer operator in scaled
//     probability space (3 exps/step). Emissions are staged into LDS by the
//     Tensor Data Mover (2D tiles: 128 rows x 48 f32, stride 192 el), double
//     buffered via TENSORcnt, with TDM LDS padding (+1 dw / 16 dw) giving a
//     204B row pitch (51 dwords, gcd(51,64)=1 -> bank-conflict-free ds reads).
//     tags/mask stream via per-tile b128 loads + one prefetch pair per tile.
// K2: per-batch fold of 64 chunk operators -> log-partition + gold score.
// K3: deterministic tree reduction -> mean.

#define B_ 2048
#define S_ 4096
#define T_ 3
#define C_ 64            // chunks per sequence
#define L_ 64            // steps per chunk (chunk c covers s in [c*64, c*64+64); s==0 skipped)
#define WS_STRIDE 12     // 9 P entries + logscale + score + count

#define BLK 128          // threads per block in K1 (= TDM tile rows)
#define TILE_STEPS 16
#define NT 4             // tiles per chunk
#define ROW_BYTES 192    // 16 steps * 3 f32 * 4B per row per tile
#define ROW_PITCH 204    // LDS pitch after TDM padding: 192 + (192/64)*4
#define BUF_BYTES (BLK * ROW_PITCH)   // 26112 per buffer

typedef unsigned int u32x4 __attribute__((ext_vector_type(4)));
typedef int          i32x8 __attribute__((ext_vector_type(8)));
typedef int          i32x4 __attribute__((ext_vector_type(4)));

// Issue one TDM 2D tile load: 128 rows x 48 f32, global row stride 768B,
// LDS padding +1 dword after every 16 dwords (row pitch 204B).
__device__ __forceinline__ void tdm_load_tile(const void* gptr, unsigned lds_addr) {
  unsigned long long ga = (unsigned long long)(size_t)gptr;
  u32x4 g0;
  g0.x = 1u;                                   // count=1 (valid descriptor)
  g0.y = lds_addr;                             // LDS byte address
  g0.z = (unsigned)ga;                         // global_addr[31:0]
  g0.w = ((unsigned)(ga >> 32) & 0x01FFFFFFu)  // global_addr[56:32]
         | 0x80000000u;                        // type=2 ("image")
  i32x8 g1;
  g1[0] = 0x00D20000;    // data_size=4B | pad_enable | pad_interval=16dw (pad=1dw)
  g1[1] = (48 << 16);    // tensor_dim0 = 48 elements (bits 79:48)
  g1[2] = (128 << 16);   // tensor_dim1 = 128 rows    (bits 111:80)
  g1[3] = (48 << 16);    // tile_dim0 = 48 elements   (bits 127:112)
  g1[4] = 128;           // tile_dim1 = 128 rows; tile_dim2 = 0
  g1[5] = 192;           // tensor_dim0_stride = 192 elements (768B)
  g1[6] = 0;
  g1[7] = 0;
  i32x4 z4 = {0, 0, 0, 0};
#if __clang_major__ >= 23
  i32x8 z8 = {0, 0, 0, 0, 0, 0, 0, 0};
  __builtin_amdgcn_tensor_load_to_lds(g0, g1, z4, z4, z8, 0);
#else
  __builtin_amdgcn_tensor_load_to_lds(g0, g1, z4, z4, 0);
#endif
}

__global__ __launch_bounds__(BLK) void crf_chunk_kernel(
    const float* __restrict__ em, const float* __restrict__ trans,
    const int* __restrict__ tags, const int* __restrict__ mask,
    float* __restrict__ ws)
{
  __shared__ __align__(16) unsigned char smem[2 * BUF_BYTES];

  const int tid = blockIdx.x * BLK + threadIdx.x;   // global chunk id
  const int c = tid & (C_ - 1);                     // chunk within sequence

  // transitions + exp(transitions) in registers
  float t00 = trans[0], t01 = trans[1], t02 = trans[2];
  float t10 = trans[3], t11 = trans[4], t12 = trans[5];
  float t20 = trans[6], t21 = trans[7], t22 = trans[8];
  float E00 = __expf(t00), E01 = __expf(t01), E02 = __expf(t02);
  float E10 = __expf(t10), E11 = __expf(t11), E12 = __expf(t12);
  float E20 = __expf(t20), E21 = __expf(t21), E22 = __expf(t22);

  // chunk transfer operator P (probability space), init = identity
  float P00 = 1.f, P01 = 0.f, P02 = 0.f;
  float P10 = 0.f, P11 = 1.f, P12 = 0.f;
  float P20 = 0.f, P21 = 0.f, P22 = 1.f;
  float logscale = 0.f, score = 0.f;
  int cnt = 0;

  const long fs_base = (long)tid * L_;              // flat step index of chunk start
  // prev tag: step before the first processed step (first processed s is 1 for c==0)
  long prev_idx = (c == 0) ? fs_base : (fs_base - 1);
  int prev = tags[prev_idx];

  const unsigned char* emb = (const unsigned char*)em;
  const long block_base = (long)blockIdx.x * (BLK * L_ * 12);  // bytes into em
  const unsigned lds_base = (unsigned)(size_t)&smem[0];

  // software pipeline: wave 0 drives the TDM, TENSORcnt completion is in-order
  if (threadIdx.x < 32) {
    tdm_load_tile(emb + block_base, lds_base);
  }
  for (int t = 0; t < NT; ++t) {
    if (threadIdx.x < 32) {
      if (t + 1 < NT) {
        tdm_load_tile(emb + block_base + (long)(t + 1) * ROW_BYTES,
                      lds_base + (unsigned)(((t + 1) & 1) * BUF_BYTES));
        __builtin_amdgcn_s_wait_tensorcnt(1);   // tile t done, tile t+1 in flight
      } else {
        __builtin_amdgcn_s_wait_tensorcnt(0);
      }
    }
    __syncthreads();   // publish buffer t&1 to all waves

    // wide b128 loads of this tile's tags/mask (64B contiguous, 16B aligned)
    const long fs_t = fs_base + t * TILE_STEPS;
    const int4* tgp = (const int4*)(tags + fs_t);
    const int4* mkp = (const int4*)(mask + fs_t);
    int4 ta = tgp[0], tb = tgp[1], tc = tgp[2], td = tgp[3];
    int4 ma = mkp[0], mb = mkp[1], mc = mkp[2], md = mkp[3];
    const int tgs[16] = {ta.x, ta.y, ta.z, ta.w, tb.x, tb.y, tb.z, tb.w,
                         tc.x, tc.y, tc.z, tc.w, td.x, td.y, td.z, td.w};
    const int mks[16] = {ma.x, ma.y, ma.z, ma.w, mb.x, mb.y, mb.z, mb.w,
                         mc.x, mc.y, mc.z, mc.w, md.x, md.y, md.z, md.w};

    // one prefetch pair per tile for the next tile's tags/mask (global_prefetch_b8)
    if (fs_t + TILE_STEPS < (long)B_ * S_) {
      __builtin_prefetch(tags + fs_t + TILE_STEPS, 0, 0);
      __builtin_prefetch(mask + fs_t + TILE_STEPS, 0, 0);
    }

    const unsigned char* rowp = smem + (t & 1) * BUF_BYTES
                              + (unsigned)threadIdx.x * ROW_PITCH;
    #pragma unroll
    for (int j = 0; j < TILE_STEPS; ++j) {
      const int s = c * L_ + t * TILE_STEPS + j;   // step within sequence

      // emissions from LDS; pad-skip offsets fold to compile-time constants
      const int rb0 = j * 12, rb1 = rb0 + 4, rb2 = rb0 + 8;
      float e0 = *(const float*)(rowp + rb0 + ((rb0 >> 6) << 2));
      float e1 = *(const float*)(rowp + rb1 + ((rb1 >> 6) << 2));
      float e2 = *(const float*)(rowp + rb2 + ((rb2 >> 6) << 2));

      const int tg = tgs[j];
      const int m  = mks[j];

      if (m > 0 && s != 0) {                      // s==0 belongs to alpha0 (K2)
        float cs = fmaxf(e0, fmaxf(e1, e2));
        float p0 = __expf(e0 - cs);
        float p1 = __expf(e1 - cs);
        float p2 = __expf(e2 - cs);
        float n00 = p0 * (P00 * E00 + P01 * E10 + P02 * E20);
        float n01 = p1 * (P00 * E01 + P01 * E11 + P02 * E21);
        float n02 = p2 * (P00 * E02 + P01 * E12 + P02 * E22);
        float n10 = p0 * (P10 * E00 + P11 * E10 + P12 * E20);
        float n11 = p1 * (P10 * E01 + P11 * E11 + P12 * E21);
        float n12 = p2 * (P10 * E02 + P11 * E12 + P12 * E22);
        float n20 = p0 * (P20 * E00 + P21 * E10 + P22 * E20);
        float n21 = p1 * (P20 * E01 + P21 * E11 + P22 * E21);
        float n22 = p2 * (P20 * E02 + P21 * E12 + P22 * E22);
        P00 = n00; P01 = n01; P02 = n02;
        P10 = n10; P11 = n11; P12 = n12;
        P20 = n20; P21 = n21; P22 = n22;
        logscale += cs;

        float et = (tg == 0) ? e0 : ((tg == 1) ? e1 : e2);
        float tp = (prev == 0) ? ((tg == 0) ? t00 : (tg == 1) ? t01 : t02)
                 : (prev == 1) ? ((tg == 0) ? t10 : (tg == 1) ? t11 : t12)
                               : ((tg == 0) ? t20 : (tg == 1) ? t21 : t22);
        score += tp + et;
        cnt++;
      }
      prev = tg;
    }

    // renormalize once per tile (16 steps): range stays in [~e^-2, ~e^19]
    {
      float mx = fmaxf(fmaxf(fmaxf(P00, P01), fmaxf(P02, P10)),
                       fmaxf(fmaxf(P11, P12), fmaxf(P20, fmaxf(P21, P22))));
      float inv = 1.0f / mx;
      P00 *= inv; P01 *= inv; P02 *= inv;
      P10 *= inv; P11 *= inv; P12 *= inv;
      P20 *= inv; P21 *= inv; P22 *= inv;
      logscale += __logf(mx);
    }
    __syncthreads();   // all reads of buffer t&1 done before it is overwritten
  }

  float* w = ws + (size_t)tid * WS_STRIDE;
  w[0] = P00; w[1] = P01; w[2] = P02;
  w[3] = P10; w[4] = P11; w[5] = P12;
  w[6] = P20; w[7] = P21; w[8] = P22;
  w[9] = logscale; w[10] = score; w[11] = (float)cnt;
}

__global__ __launch_bounds__(256) void crf_combine_kernel(
    const float* __restrict__ em, const float* __restrict__ startt,
    const float* __restrict__ endt, const int* __restrict__ tags,
    const int* __restrict__ mask, const float* __restrict__ ws,
    float* __restrict__ loss)
{
  int b = blockIdx.x * blockDim.x + threadIdx.x;
  if (b >= B_) return;
  long base = (long)b * S_;
  const float* e0p = em + base * 3;

  float a0 = startt[0] + e0p[0];
  float a1 = startt[1] + e0p[1];
  float a2 = startt[2] + e0p[2];

  float score_sum = 0.f;
  int cnt = (mask[base] > 0) ? 1 : 0;

  const float* w = ws + (size_t)b * C_ * WS_STRIDE;
  for (int c = 0; c < C_; ++c, w += WS_STRIDE) {
    float amax = fmaxf(a0, fmaxf(a1, a2));
    float x0 = __expf(a0 - amax);
    float x1 = __expf(a1 - amax);
    float x2 = __expf(a2 - amax);
    float q0 = x0 * w[0] + x1 * w[3] + x2 * w[6];
    float q1 = x0 * w[1] + x1 * w[4] + x2 * w[7];
    float q2 = x0 * w[2] + x1 * w[5] + x2 * w[8];
    float ls = amax + w[9];
    a0 = ls + __logf(q0);
    a1 = ls + __logf(q1);
    a2 = ls + __logf(q2);
    score_sum += w[10];
    cnt += (int)w[11];
  }

  float z0 = a0 + endt[0], z1 = a1 + endt[1], z2 = a2 + endt[2];
  float zm = fmaxf(z0, fmaxf(z1, z2));
  float lp = zm + __logf(__expf(z0 - zm) + __expf(z1 - zm) + __expf(z2 - zm));

  int tg0 = tags[base];
  float score = startt[tg0] + e0p[tg0] + score_sum;
  int li = cnt - 1;
  if (li < 0) li = 0;
  if (li >= S_) li = S_ - 1;
  score += endt[tags[base + li]];

  loss[b] = lp - score;
}

__global__ __launch_bounds__(256) void crf_reduce_kernel(
    const float* __restrict__ loss, float* __restrict__ out)
{
  __shared__ float sm[256];
  float s = 0.f;
  for (int i = threadIdx.x; i < B_; i += 256) s += loss[i];
  sm[threadIdx.x] = s;
  __syncthreads();
  for (int off = 128; off > 0; off >>= 1) {
    if ((int)threadIdx.x < off) sm[threadIdx.x] += sm[threadIdx.x + off];
    __syncthreads();
  }
  if (threadIdx.x == 0) out[0] = sm[0] * (1.0f / (float)B_);
}

extern "C" void kernel_launch(void* const* d_in, const int* in_sizes, int n_in,
                              void* d_out, int out_size, void* d_ws, size_t ws_size,
                              hipStream_t stream) {
  const float* em     = (const float*)d_in[0];  // (B, S, T)
  const float* trans  = (const float*)d_in[1];  // (T, T)
  const float* startt = (const float*)d_in[2];  // (T,)
  const float* endt   = (const float*)d_in[3];  // (T,)
  const int*   tags   = (const int*)d_in[4];    // (B, S)
  const int*   mask   = (const int*)d_in[5];    // (B, S)
  float* out = (float*)d_out;

  float* wsP  = (float*)d_ws;                       // B*C*12 floats (~6.3 MB)
  float* loss = wsP + (size_t)B_ * C_ * WS_STRIDE;  // B floats

  crf_chunk_kernel<<<(B_ * C_) / BLK, BLK, 0, stream>>>(em, trans, tags, mask, wsP);
  crf_combine_kernel<<<B_ / 256, 256, 0, stream>>>(em, startt, endt, tags, mask, wsP, loss);
  crf_reduce_kernel<<<1, 256, 0, stream>>>(loss, out);
}